// QuadtreeMRF_6751688589409
// MI455X (gfx1250) — compile-verified
//
#include <hip/hip_runtime.h>
#include <math.h>

typedef __attribute__((ext_vector_type(2))) float v2f;
typedef __attribute__((ext_vector_type(8))) float v8f;

#define B_    16
#define CL_   256
#define H_    128
#define G_    16
#define LEAF_ 8
#define HID_  128
#define NC_   6
#define BPIT_ 5
#define EPSV  1e-10f

// ---------------------------------------------------------------------------
// Kernel 1: 8x8 average pooling of cvae_latent (B,CL,H,H) -> pooled laid out
// as (pos, CL) with pos = b*256 + gi*16 + gj  (GEMM A-matrix, row major).
// One block per (b, c, gi): the 8-row stripe is 4 KB contiguous memory.
// ---------------------------------------------------------------------------
__global__ void __launch_bounds__(256)
pool_kernel(const float* __restrict__ cvae, float* __restrict__ pooled) {
    __shared__ float acc[G_];
    const int t   = threadIdx.x;
    const int blk = blockIdx.x;            // b*CL*G + c*G + gi
    const int gi  = blk % G_;
    const int bc  = blk / G_;              // b*CL + c
    const int c   = bc % CL_;
    const int b   = bc / CL_;

    if (t < G_) acc[t] = 0.0f;
    __syncthreads();

    // stripe base: rows [gi*8, gi*8+8) of image (b,c) -> 1024 contiguous floats
    const float4* src = (const float4*)(cvae + ((size_t)bc * H_ + gi * LEAF_) * H_);
    float4 v = src[t];                     // thread t: flat floats [4t, 4t+4)
    float  s = v.x + v.y + v.z + v.w;
    s += __shfl_xor(s, 1, 32);             // pair (t, t^1) covers one 8-col block row
    const int col = (t * 4) & (H_ - 1);    // column of first element
    const int gj  = col >> 3;              // 8-col block index
    if ((t & 1) == 0) atomicAdd(&acc[gj], s);   // ds_add_f32
    __syncthreads();

    if (t < G_) {
        const size_t pos = ((size_t)(b * G_ + gi)) * G_ + t;   // gj = t
        pooled[pos * CL_ + c] = acc[t] * (1.0f / 64.0f);
    }
}

// ---------------------------------------------------------------------------
// Helpers for the fused kernel
// ---------------------------------------------------------------------------
__device__ __forceinline__ void softmax6(const float* __restrict__ x,
                                         float* __restrict__ y) {
    float mx = x[0];
#pragma unroll
    for (int i = 1; i < NC_; ++i) mx = fmaxf(mx, x[i]);
    float s = 0.0f;
#pragma unroll
    for (int i = 0; i < NC_; ++i) { y[i] = __expf(x[i] - mx); s += y[i]; }
    const float inv = 1.0f / s;
#pragma unroll
    for (int i = 0; i < NC_; ++i) y[i] *= inv;
}

// ---------------------------------------------------------------------------
// Kernel 2: fused MLP (f32 WMMA) + belief propagation + argmax + 8x8 upsample.
// One 256-thread block (8 waves) per (b, gi) row of 16 positions.
//   GEMM1: h[16,128] = relu(A[16,256] * w1^T + b1) via V_WMMA_F32_16X16X4_F32
//   wave w computes hid tile [16w, 16w+16) over all 16 positions.
// ---------------------------------------------------------------------------
__global__ void __launch_bounds__(256)
mrf_kernel(const float* __restrict__ pooled, const float* __restrict__ pw,
           const float* __restrict__ w1, const float* __restrict__ b1,
           const float* __restrict__ w2, const float* __restrict__ b2,
           float* __restrict__ out) {
    __shared__ float A_lds[16 * CL_];      // 16 KB : pooled tile, row major
    __shared__ float h_lds[16 * HID_];     //  8 KB : hidden activations
    __shared__ float pred_lds[G_];

    const int t  = threadIdx.x;
    const int wg = blockIdx.x;             // b*16 + gi
    const int gi = wg % G_;
    const int b  = wg / G_;

    // ---- stage A tile: positions [wg*16, wg*16+16), 4096 contiguous floats
    {
        const float4* src = (const float4*)(pooled + (size_t)wg * 16 * CL_);
        float4* dst = (float4*)A_lds;
#pragma unroll
        for (int i = 0; i < 4; ++i) dst[t + 256 * i] = src[t + 256 * i];
    }
    __syncthreads();

    const int lane   = t & 31;
    const int wave   = t >> 5;             // 0..7
    const int n_base = wave * 16;          // hid tile start
    const int m16    = lane & 15;
    const int koff   = (lane >> 4) << 1;   // 0 or 2 : K-pair selector (ISA A/B layout)

    // ---- GEMM1 with chained f32 WMMA (K accumulated in C operand)
    v8f c8 = {};
    const float* arow  = A_lds + m16 * CL_;                 // A row M = lane&15
    const float* w1row = w1 + (size_t)(n_base + m16) * CL_; // B col N = lane&15 (w1 is [HID][CL])
#pragma unroll 8
    for (int k = 0; k < CL_; k += 4) {
        v2f a  = *(const v2f*)(arow  + k + koff);   // A[m][k+koff], A[m][k+koff+1]
        v2f bb = *(const v2f*)(w1row + k + koff);   // B[k+koff][n], B[k+koff+1][n]
        c8 = __builtin_amdgcn_wmma_f32_16x16x4_f32(
                 false, a, false, bb, (short)0, c8, false, false);
    }

    // ---- bias + relu, scatter C tile to LDS (m = r + 8*(lane>>4), n = n_base + (lane&15))
    {
        const int n    = n_base + m16;
        const int mhi  = (lane >> 4) << 3;
        const float bias = b1[n];
#pragma unroll
        for (int r = 0; r < 8; ++r) {
            float hv = c8[r] + bias;
            h_lds[(mhi + r) * HID_ + n] = hv > 0.0f ? hv : 0.0f;
        }
    }
    __syncthreads();

    // ---- per-position: 128->6 projection, BP iterations, argmax (lanes 0..15)
    if (t < G_) {
        const int gj = t;
        float u[NC_];
#pragma unroll
        for (int c = 0; c < NC_; ++c) {
            float s = b2[c];
            const float* w2r = w2 + c * HID_;
            const float* hr  = h_lds + gj * HID_;
            for (int h = 0; h < HID_; ++h) s += hr[h] * w2r[h];
            u[c] = s;
        }

        float P[NC_ * NC_];
#pragma unroll
        for (int i = 0; i < NC_ * NC_; ++i) P[i] = pw[i];

        const int edge_i = (gi == 0) + (gi == G_ - 1);
        const int edge_j = (gj == 0) + (gj == G_ - 1);
        const int nnb    = 4 - edge_i - edge_j;

        float bel[NC_];
        softmax6(u, bel);
        for (int it = 0; it < BPIT_; ++it) {
            float acc[NC_], lsum[NC_];
#pragma unroll
            for (int c = 0; c < NC_; ++c) { acc[c] = bel[c]; lsum[c] = 0.0f; }
#pragma unroll
            for (int p = 0; p < 4; ++p) {
                float mm[NC_]; float s = 0.0f;
#pragma unroll
                for (int c = 0; c < NC_; ++c) {
                    float m = 0.0f;
#pragma unroll
                    for (int k = 0; k < NC_; ++k) m += P[c * NC_ + k] * acc[k];
                    mm[c] = m; s += m;
                }
                const float inv = 1.0f / s;
#pragma unroll
                for (int c = 0; c < NC_; ++c) mm[c] *= inv;
                if (nnb > p) {
#pragma unroll
                    for (int c = 0; c < NC_; ++c) lsum[c] += __logf(mm[c] + EPSV);
                }
#pragma unroll
                for (int c = 0; c < NC_; ++c) acc[c] *= mm[c];
            }
            float z[NC_];
#pragma unroll
            for (int c = 0; c < NC_; ++c) z[c] = u[c] + lsum[c];
            softmax6(z, bel);
        }

        int best = 0; float bv = bel[0];
#pragma unroll
        for (int c = 1; c < NC_; ++c) if (bel[c] > bv) { bv = bel[c]; best = c; }
        pred_lds[gj] = (float)best;
    }
    __syncthreads();

    // ---- 8x8 upsample: 8 rows x 128 cols per block, coalesced float4 stores
    {
        const int r  = t >> 5;             // row in stripe 0..7
        const int cq = t & 31;             // float4 column index 0..31
        const float v = pred_lds[cq >> 1]; // gj = (cq*4)/8
        float4 o4 = {v, v, v, v};
        float4* op = (float4*)(out + ((size_t)(b * H_) + gi * LEAF_ + r) * H_);
        op[cq] = o4;
    }
}

// ---------------------------------------------------------------------------
// Launch: inputs are (features, cvae_latent, pairwise_weights, w1, b1, w2, b2)
// features is unused by the reference. d_ws holds the 4 MB pooled tensor.
// ---------------------------------------------------------------------------
extern "C" void kernel_launch(void* const* d_in, const int* in_sizes, int n_in,
                              void* d_out, int out_size, void* d_ws, size_t ws_size,
                              hipStream_t stream) {
    (void)in_sizes; (void)n_in; (void)out_size; (void)ws_size;
    const float* cvae = (const float*)d_in[1];
    const float* pw   = (const float*)d_in[2];
    const float* w1   = (const float*)d_in[3];
    const float* b1   = (const float*)d_in[4];
    const float* w2   = (const float*)d_in[5];
    const float* b2   = (const float*)d_in[6];
    float* pooled = (float*)d_ws;                    // (4096, 256) f32 = 4 MB
    float* out    = (float*)d_out;                   // (16, 128, 128) f32

    pool_kernel<<<B_ * CL_ * G_, 256, 0, stream>>>(cvae, pooled);
    mrf_kernel <<<B_ * G_,       256, 0, stream>>>(pooled, pw, w1, b1, w2, b2, out);
}